// HierarchicalPoolingLayer_65841848648304
// MI455X (gfx1250) — compile-verified
//
#include <hip/hip_runtime.h>
#include <math.h>

#define N0 100000
#define DD 1443
#define K3 12500
#define ROW_CHUNKS 64
#define NSTEP ((DD + 3) / 4)   // 361 K-steps of 4 (last is zero-padded tail)

// workspace layout, in float (4-byte) units
#define OFF_D     0          // 3 * N0 dots: d1,d2,d3
#define OFF_IDXA  300000     // N0 ints
#define OFF_COEFA 400000     // N0 floats
#define OFF_IDXB  500000     // N0 ints
#define OFF_COEFB 600000     // N0 floats
#define OFF_SCORE 700000     // N0 floats
#define OFF_PART  800000     // ROW_CHUNKS * DD floats (92352)
#define OFF_WB    900000     // NSTEP * 32 * 2 floats (23104) packed B operands

typedef __attribute__((ext_vector_type(2))) float v2f;
typedef __attribute__((ext_vector_type(8))) float v8f;

// ---------------------------------------------------------------------------
// Phase 0: pack the WMMA B operand per (kstep, lane) so the hot loop has zero
// divergence. B is 4x16 (KxN): lanes 0-15 hold K{0,1} for N=lane, lanes 16-31
// hold K{2,3} for N=lane-16. Columns 0..2 = w1,w2,w3; rest zero; tail zero-pad.
// ---------------------------------------------------------------------------
__global__ void pack_wb_kernel(const float* __restrict__ w1,
                               const float* __restrict__ w2,
                               const float* __restrict__ w3,
                               float* __restrict__ wb) {
  int t = blockIdx.x * blockDim.x + threadIdx.x;   // one thread per (kstep,lane)
  if (t >= NSTEP * 32) return;
  int kstep = t >> 5, lane = t & 31;
  int half = lane >> 4, m = lane & 15;
  int k0 = kstep * 4 + 2 * half;
  const float* wv = (m == 0) ? w1 : (m == 1) ? w2 : w3;
  float bx = (m < 3 && k0     < DD) ? wv[k0]     : 0.f;
  float by = (m < 3 && k0 + 1 < DD) ? wv[k0 + 1] : 0.f;
  wb[t * 2 + 0] = bx;
  wb[t * 2 + 1] = by;
}

// ---------------------------------------------------------------------------
// Phase 1: d_j[row] = x[row] . w_j via V_WMMA_F32_16X16X4_F32.
// One wave = 16 rows. Inner loop: 2 unconditional b64 loads + 1 WMMA.
// ---------------------------------------------------------------------------
__global__ void dots_wmma_kernel(const float* __restrict__ x,
                                 const float* __restrict__ wb,
                                 float* __restrict__ dout) {
  const int wave = (blockIdx.x * blockDim.x + threadIdx.x) >> 5;
  const int lane = threadIdx.x & 31;
  const int row_base = wave << 4;
  if (row_base >= N0) return;                 // wave-uniform exit (EXEC stays full)
  const int half = lane >> 4;                 // 0 -> K{k,k+1}, 1 -> K{k+2,k+3}
  const int m    = lane & 15;                 // row within tile / B column
  const int koff = 2 * half;
  const float* xrow = x + (size_t)(row_base + m) * DD;
  const v2f*  wbv  = (const v2f*)wb;          // [kstep*32 + lane]

  v8f c = {0.f, 0.f, 0.f, 0.f, 0.f, 0.f, 0.f, 0.f};
  int k = 0, kstep = 0;
  for (; k + 4 <= DD; k += 4, ++kstep) {
    v2f a;
    a.x = xrow[k + koff];
    a.y = xrow[k + koff + 1];
    v2f b = wbv[kstep * 32 + lane];
    c = __builtin_amdgcn_wmma_f32_16x16x4_f32(false, a, false, b, (short)0, c,
                                              false, false);
  }
  { // tail: DD % 4 == 3; B side already zero-padded in the table
    v2f a = {0.f, 0.f};
    int k0 = k + koff;
    if (k0 < DD)     a.x = xrow[k0];
    if (k0 + 1 < DD) a.y = xrow[k0 + 1];
    v2f b = wbv[kstep * 32 + lane];
    c = __builtin_amdgcn_wmma_f32_16x16x4_f32(false, a, false, b, (short)0, c,
                                              false, false);
  }
  if (m < 3) {
    float* dst = dout + (size_t)m * N0 + row_base + half * 8;
#pragma unroll
    for (int r = 0; r < 8; ++r) dst[r] = c[r];
  }
}

// ---------------------------------------------------------------------------
// Phase 2: cascaded exact top-k selection, single 1024-thread workgroup.
// ---------------------------------------------------------------------------
__device__ __forceinline__ unsigned sortable(float f) {
  unsigned u = __float_as_uint(f);
  return (u & 0x80000000u) ? ~u : (u | 0x80000000u);  // monotone float order
}

__device__ int block_scan_excl(int v, int tid, int* sh, int& total) {
  sh[tid] = v;
  __syncthreads();
  for (int off = 1; off < 1024; off <<= 1) {
    int t = (tid >= off) ? sh[tid - off] : 0;
    __syncthreads();
    sh[tid] += t;
    __syncthreads();
  }
  int incl = sh[tid];
  total = sh[1023];
  __syncthreads();
  return incl - v;
}

__global__ __launch_bounds__(1024)
void select_kernel(const float* __restrict__ w1, const float* __restrict__ w2,
                   const float* __restrict__ w3, float* __restrict__ ws) {
  __shared__ float redf[1024];
  __shared__ int   scan[1024];
  __shared__ int   hist[256];
  __shared__ float sh_inv[3];
  __shared__ int   sh_bin, sh_acc;
  const int tid = threadIdx.x;

  // 1/||w_j||
  const float* wv[3] = {w1, w2, w3};
  for (int j = 0; j < 3; ++j) {
    float s = 0.f;
    for (int i = tid; i < DD; i += 1024) { float v = wv[j][i]; s += v * v; }
    redf[tid] = s; __syncthreads();
    for (int off = 512; off > 0; off >>= 1) {
      if (tid < off) redf[tid] += redf[tid + off];
      __syncthreads();
    }
    if (tid == 0) sh_inv[j] = 1.0f / sqrtf(redf[0]);
    __syncthreads();
  }

  float* score = ws + OFF_SCORE;
  int n = N0;
  for (int l = 0; l < 3; ++l) {
    const float* dW = ws + OFF_D + (size_t)l * N0;
    const int*   idx_in;  const float* coef_in;
    int*         idx_out; float*       coef_out;
    if (l == 0) { idx_in = nullptr;                  coef_in = nullptr;
                  idx_out = (int*)(ws + OFF_IDXA);   coef_out = ws + OFF_COEFA; }
    else if (l == 1) { idx_in = (const int*)(ws + OFF_IDXA); coef_in = ws + OFF_COEFA;
                       idx_out = (int*)(ws + OFF_IDXB);      coef_out = ws + OFF_COEFB; }
    else { idx_in = (const int*)(ws + OFF_IDXB); coef_in = ws + OFF_COEFB;
           idx_out = (int*)(ws + OFF_IDXA);      coef_out = ws + OFF_COEFA; }
    const float inv = sh_inv[l];
    const int kk = (l == 0) ? 50000 : (l == 1) ? 25000 : 12500;

    // scores for this level
    for (int i = tid; i < n; i += 1024) {
      int   orig = idx_in ? idx_in[i] : i;
      float cin  = coef_in ? coef_in[i] : 1.0f;
      score[i] = tanhf(cin * dW[orig] * inv);
    }
    __syncthreads();

    // exact radix-select of the k-th largest key
    unsigned prefix = 0; int need = kk;
    for (int p = 3; p >= 0; --p) {
      if (tid < 256) hist[tid] = 0;
      __syncthreads();
      for (int i = tid; i < n; i += 1024) {
        unsigned u = sortable(score[i]);
        bool ok = (p == 3) || ((u >> ((p + 1) * 8)) == prefix);
        if (ok) atomicAdd(&hist[(u >> (p * 8)) & 255], 1);
      }
      __syncthreads();
      if (tid == 0) {
        int acc = 0, b = 255;
        for (; b > 0; --b) { if (acc + hist[b] >= need) break; acc += hist[b]; }
        sh_bin = b; sh_acc = acc;
      }
      __syncthreads();
      prefix = (prefix << 8) | (unsigned)sh_bin;
      need  -= sh_acc;
      __syncthreads();
    }
    const unsigned T = prefix;
    const int tie_need = need;   // #elements equal to T to take (lowest index first)

    // index-ordered compaction (deterministic; matches jax top_k tie-break)
    int out_base = 0, tie_used = 0;
    for (int base = 0; base < n; base += 1024) {
      int i = base + tid;
      int fgt = 0, feq = 0;
      if (i < n) {
        unsigned u = sortable(score[i]);
        fgt = (u > T) ? 1 : 0;
        feq = (u == T) ? 1 : 0;
      }
      int totEQ;  int posEQ = block_scan_excl(feq, tid, scan, totEQ);
      int sel = fgt | (feq && (tie_used + posEQ) < tie_need);
      int totSel; int pos  = block_scan_excl(sel, tid, scan, totSel);
      if (sel) {
        int slot = out_base + pos;
        int   orig = idx_in ? idx_in[i] : i;
        float cin  = coef_in ? coef_in[i] : 1.0f;
        idx_out[slot]  = orig;
        coef_out[slot] = cin * score[i];
      }
      out_base += totSel;
      int rem  = tie_need - tie_used;
      int used = totEQ < rem ? totEQ : rem;
      tie_used += (used > 0) ? used : 0;
      __syncthreads();
    }
    n = kk;
    __syncthreads();
  }
  // final 12500 (idx, coef) pairs end up in buffers A
}

// ---------------------------------------------------------------------------
// Phase 3: deterministic weighted gather-mean over 12500 rows.
// ---------------------------------------------------------------------------
__global__ void gather_partial_kernel(const float* __restrict__ x,
                                      const float* __restrict__ ws,
                                      float* __restrict__ part) {
  const int col   = blockIdx.x * blockDim.x + threadIdx.x;
  const int chunk = blockIdx.y;
  if (col >= DD) return;
  const int r0 = (chunk * K3) / ROW_CHUNKS;
  const int r1 = ((chunk + 1) * K3) / ROW_CHUNKS;
  const int*   idx  = (const int*)(ws + OFF_IDXA);
  const float* coef = ws + OFF_COEFA;
  float acc = 0.f;
  for (int r = r0; r < r1; ++r)
    acc += coef[r] * x[(size_t)idx[r] * DD + col];
  part[(size_t)chunk * DD + col] = acc;
}

__global__ void final_reduce_kernel(const float* __restrict__ part,
                                    float* __restrict__ out) {
  int col = blockIdx.x * blockDim.x + threadIdx.x;
  if (col >= DD) return;
  float s = 0.f;
  for (int c = 0; c < ROW_CHUNKS; ++c) s += part[(size_t)c * DD + col];
  out[col] = s * (1.0f / (float)K3);
}

// ---------------------------------------------------------------------------
extern "C" void kernel_launch(void* const* d_in, const int* in_sizes, int n_in,
                              void* d_out, int out_size, void* d_ws, size_t ws_size,
                              hipStream_t stream) {
  const float* x  = (const float*)d_in[0];
  // d_in[1] = edge_index (int64), d_in[2] = edge_attr: dead inputs
  const float* w1 = (const float*)d_in[3];
  const float* w2 = (const float*)d_in[4];
  const float* w3 = (const float*)d_in[5];
  float* ws  = (float*)d_ws;
  float* out = (float*)d_out;

  pack_wb_kernel<<<(NSTEP * 32 + 255) / 256, 256, 0, stream>>>(w1, w2, w3,
                                                               ws + OFF_WB);

  const int waves  = N0 / 16;                       // 6250 waves, 16 rows each
  const int blocks = (waves * 32 + 255) / 256;      // 782 blocks of 8 waves
  dots_wmma_kernel<<<blocks, 256, 0, stream>>>(x, ws + OFF_WB, ws + OFF_D);

  select_kernel<<<1, 1024, 0, stream>>>(w1, w2, w3, ws);

  dim3 gC((DD + 255) / 256, ROW_CHUNKS);
  gather_partial_kernel<<<gC, 256, 0, stream>>>(x, ws, ws + OFF_PART);

  final_reduce_kernel<<<(DD + 255) / 256, 256, 0, stream>>>(ws + OFF_PART, out);
}